// LatentEdgeFilter_85358180041297
// MI455X (gfx1250) — compile-verified
//
#include <hip/hip_runtime.h>
#include <hip/hip_bf16.h>
#include <stdint.h>

// Problem constants (reference: x = [32, 4, 512, 512] f32, fixed k3/k5 stencils)
#define NB   32
#define NCH  4
#define IH   512
#define IW   512

// Tiling: 128x16 output tile per workgroup, 128 threads (4 waves, wave32),
// each thread computes a 4x4 output block. Halo = 2 (5x5 stencil).
// LDS tile has 4 extra cols on the left so every b128 chunk is 16B aligned
// both in LDS and in global memory (tile col 0 <-> global col tx*128 - 4).
#define TW     128
#define TH     16
#define LROWS  (TH + 4)    // 20 rows  (vertical halo 2+2)
#define LPITCH (TW + 8)    // 136 floats = 544 B per row (16B multiple), 34 chunks
#define NTHR   128

typedef int v4i __attribute__((ext_vector_type(4)));
typedef __attribute__((address_space(1))) v4i* gas_v4i_ptr;   // global (AS1) int4*
typedef __attribute__((address_space(3))) v4i* las_v4i_ptr;   // LDS    (AS3) int4*

// ---------------- CDNA5 async global->LDS helpers ----------------
__device__ __forceinline__ void async_load_b128(void* lds_dst, const void* gsrc) {
#if __has_builtin(__builtin_amdgcn_global_load_async_to_lds_b128)
    __builtin_amdgcn_global_load_async_to_lds_b128(
        (gas_v4i_ptr)(uintptr_t)gsrc,      // global source (AS1 int4*)
        (las_v4i_ptr)lds_dst,              // LDS destination (AS3 int4*)
        0, 0);
#else
    unsigned ldsoff = (unsigned)(uintptr_t)lds_dst;   // LDS aperture: low 32 bits = offset
    asm volatile("global_load_async_to_lds_b128 %0, %1, off"
                 :: "v"(ldsoff), "v"(gsrc) : "memory");
#endif
}

__device__ __forceinline__ void wait_async0() {
#if __has_builtin(__builtin_amdgcn_s_wait_asynccnt)
    __builtin_amdgcn_s_wait_asynccnt(0);
#else
    asm volatile("s_wait_asynccnt 0" ::: "memory");
#endif
}

// ---------------- kernels (edge kernel first so it leads the disassembly) ----

__global__ __launch_bounds__(NTHR)
void lef_edge_kernel(const float* __restrict__ x,
                     float* __restrict__ out,
                     unsigned* __restrict__ vmin,
                     unsigned* __restrict__ vmax,
                     const float* __restrict__ zero16) {
    __shared__ __align__(16) float tile[NCH][LROWS][LPITCH];   // 43,520 B
    __shared__ float smin[NTHR], smax[NTHR];

    const int tid     = threadIdx.x;
    const int tilesX  = IW / TW;            // 4
    const int tilesY  = IH / TH;            // 32
    const int tpi     = tilesX * tilesY;    // 128 tiles per image
    const int b       = blockIdx.x / tpi;
    const int t       = blockIdx.x % tpi;
    const int ty      = t / tilesX;
    const int tx      = t % tilesX;

    const int gx0 = tx * TW - 4;            // global col of LDS col 0 (16B aligned)
    const int gy0 = ty * TH - 2;            // global row of LDS row 0

    // ---- stage halo tile into LDS via async b128 copies ----
    // Structural mapping, no div/mod: wave w <-> channel w (4 waves, 4 channels),
    // loop counter <-> tile row (20), lane <-> b128 chunk (lanes 0-1 also take
    // chunks 32-33 of the 34-chunk row).
    const int wave = tid >> 5;              // channel
    const int lane = tid & 31;
    const float* xb = x + (size_t)b * NCH * IH * IW + (size_t)wave * IH * IW;
    const int gxa = gx0 + 4 * lane;               // chunk 'lane'
    const int gxb = gx0 + 4 * (32 + lane);        // chunk 32+lane (lanes 0-1 only)
    const bool cola = (gxa >= 0) & (gxa < IW);    // chunks never straddle the edge
    const bool colb = (gxb >= 0) & (gxb < IW);
#pragma unroll 4
    for (int r = 0; r < LROWS; ++r) {
        const int gy = gy0 + r;
        const bool rowok = (gy >= 0) & (gy < IH);
        const float* rowbase = xb + (size_t)gy * IW;
        const float* s0 = (rowok & cola) ? (rowbase + gxa) : zero16;
        async_load_b128(&tile[wave][r][4 * lane], s0);
        if (lane < 2) {
            const float* s1 = (rowok & colb) ? (rowbase + gxb) : zero16;
            async_load_b128(&tile[wave][r][4 * (32 + lane)], s1);
        }
    }
    wait_async0();          // my wave's async LDS writes complete
    __syncthreads();        // everyone's complete

    // ---- compute: 4x4 outputs per thread, channel-max fused ----
    const int cx = (tid & 31) * 4;          // 0..124 : output col base (LDS col cx+4)
    const int ry = (tid >> 5) * 4;          // 0..12  : output row base (LDS row ry+2)

    float acc[4][4];
#pragma unroll
    for (int r = 0; r < 4; ++r)
#pragma unroll
        for (int j = 0; j < 4; ++j) acc[r][j] = 0.0f;   // responses are >= 0

#pragma unroll
    for (int c = 0; c < NCH; ++c) {
        // register window: wv[i][k] = tile[c][ry+i][cx+k], i in 0..7, k in 0..11
        float wv[8][12];
#pragma unroll
        for (int rr = 1; rr <= 6; ++rr) {
            const float4 a = *(const float4*)&tile[c][ry + rr][cx];
            const float4 m = *(const float4*)&tile[c][ry + rr][cx + 4];
            const float4 d = *(const float4*)&tile[c][ry + rr][cx + 8];
            wv[rr][0] = a.x; wv[rr][1] = a.y; wv[rr][2]  = a.z; wv[rr][3]  = a.w;
            wv[rr][4] = m.x; wv[rr][5] = m.y; wv[rr][6]  = m.z; wv[rr][7]  = m.w;
            wv[rr][8] = d.x; wv[rr][9] = d.y; wv[rr][10] = d.z; wv[rr][11] = d.w;
        }
        {
            const float4 tq = *(const float4*)&tile[c][ry][cx + 4];
            wv[0][4] = tq.x; wv[0][5] = tq.y; wv[0][6] = tq.z; wv[0][7] = tq.w;
            const float4 bq = *(const float4*)&tile[c][ry + 7][cx + 4];
            wv[7][4] = bq.x; wv[7][5] = bq.y; wv[7][6] = bq.z; wv[7][7] = bq.w;
        }
#pragma unroll
        for (int r = 0; r < 4; ++r) {
#pragma unroll
            for (int j = 0; j < 4; ++j) {
                const float cc = wv[r + 2][4 + j];
                const float nn = wv[r + 1][4 + j];
                const float ss = wv[r + 3][4 + j];
                const float ww = wv[r + 2][3 + j];
                const float ee = wv[r + 2][5 + j];
                const float s1 = (nn + ss) + (ww + ee);
                // 3x3 Laplacian: n+s+w+e - 4c
                const float e3 = s1 - 4.0f * cc;
                const float dg = (wv[r + 1][3 + j] + wv[r + 1][5 + j])
                               + (wv[r + 3][3 + j] + wv[r + 3][5 + j]);
                const float fr = (wv[r][4 + j] + wv[r + 4][4 + j])
                               + (wv[r + 2][2 + j] + wv[r + 2][6 + j]);
                // 5x5 LoG: 16c - 2*(n+s+w+e) - diagonals - far taps
                const float e5 = 16.0f * cc - 2.0f * s1 - dg - fr;
                const float resp = fmaxf(fabsf(e3), fabsf(e5));
                acc[r][j] = fmaxf(acc[r][j], resp);
            }
        }
    }

    // ---- store edge map + block min/max reduction ----
    float tmin = 3.4e38f, tmax = 0.0f;
#pragma unroll
    for (int r = 0; r < 4; ++r) {
        const int gy = ty * TH + ry + r;
        float4 o4;
        o4.x = acc[r][0]; o4.y = acc[r][1]; o4.z = acc[r][2]; o4.w = acc[r][3];
        *(float4*)&out[((size_t)b * IH + gy) * IW + tx * TW + cx] = o4;
        tmin = fminf(tmin, fminf(fminf(o4.x, o4.y), fminf(o4.z, o4.w)));
        tmax = fmaxf(tmax, fmaxf(fmaxf(o4.x, o4.y), fmaxf(o4.z, o4.w)));
    }
    smin[tid] = tmin;
    smax[tid] = tmax;
    __syncthreads();
    for (int off = NTHR / 2; off > 0; off >>= 1) {
        if (tid < off) {
            smin[tid] = fminf(smin[tid], smin[tid + off]);
            smax[tid] = fmaxf(smax[tid], smax[tid + off]);
        }
        __syncthreads();
    }
    if (tid == 0) {
        // all values >= 0 -> IEEE bit pattern order == float order
        atomicMin(&vmin[b], __float_as_uint(smin[0]));
        atomicMax(&vmax[b], __float_as_uint(smax[0]));
    }
}

// init d_ws: per-batch min/max (as orderable u32, values are >= 0) + 16B zero chunk
__global__ void lef_init_kernel(unsigned* __restrict__ vmin,
                                unsigned* __restrict__ vmax,
                                float* __restrict__ zero16) {
    const int i = threadIdx.x;
    if (i < NB) { vmin[i] = 0x7F800000u; /* +inf */ vmax[i] = 0u; }
    if (i < 4)  zero16[i] = 0.0f;
}

__global__ __launch_bounds__(256)
void lef_norm_kernel(float* __restrict__ out,
                     const unsigned* __restrict__ vmin,
                     const unsigned* __restrict__ vmax) {
    const int i = blockIdx.x * 256 + threadIdx.x;   // float4 index
    const int b = i >> 16;                          // 65536 float4 per image
    const float lo = __uint_as_float(vmin[b]);
    const float hi = __uint_as_float(vmax[b]);
    const float sc = 1.0f / (hi - lo + 1e-6f);
    float4 v = *((const float4*)out + i);
    v.x = (v.x - lo) * sc;
    v.y = (v.y - lo) * sc;
    v.z = (v.z - lo) * sc;
    v.w = (v.w - lo) * sc;
    *((float4*)out + i) = v;
}

// ---------------- launch ----------------
extern "C" void kernel_launch(void* const* d_in, const int* in_sizes, int n_in,
                              void* d_out, int out_size, void* d_ws, size_t ws_size,
                              hipStream_t stream) {
    (void)in_sizes; (void)n_in; (void)out_size; (void)ws_size;
    const float* x = (const float*)d_in[0];   // [32,4,512,512]; k3/k5 constants are baked in
    float* out = (float*)d_out;               // [32,1,512,512] (staged raw, normalized in place)

    unsigned* vmin   = (unsigned*)d_ws;                 // 32 u32
    unsigned* vmax   = vmin + NB;                       // 32 u32
    float*    zero16 = (float*)((char*)d_ws + 256);     // 16B zero chunk, 16B aligned

    lef_init_kernel<<<1, 64, 0, stream>>>(vmin, vmax, zero16);

    const int tilesPerImage = (IW / TW) * (IH / TH);    // 128
    lef_edge_kernel<<<NB * tilesPerImage, NTHR, 0, stream>>>(x, out, vmin, vmax, zero16);

    const int nvec4 = NB * IH * IW / 4;                 // 2,097,152
    lef_norm_kernel<<<nvec4 / 256, 256, 0, stream>>>(out, vmin, vmax);
}